// QuantumDenseLayer_4312147165378
// MI455X (gfx1250) — compile-verified
//
#include <hip/hip_runtime.h>
#include <hip/hip_bf16.h>
#include <math.h>

// ---------------------------------------------------------------------------
// QuantumDenseLayer for MI455X (gfx1250).
//   state row (4096 c64) viewed as 64x64 matrix S; rotation layer == A*S*B^T
//   with A = U0..U5 kron, B = U6..U11 kron (64x64 complex, built once).
//   CNOT cascade == precomputed index permutation. f32 WMMA 16x16x4 performs
//   the six real 64x64x64 matmuls per row (initial state is real -> phase A
//   needs only 2 matmuls, phase B needs 4).
//   Both A and B are stored row-major [out][in] so every WMMA fragment's
//   .x/.y pair is contiguous -> global_load_b64 on both phases.
// ---------------------------------------------------------------------------

typedef float v2f __attribute__((ext_vector_type(2)));
typedef float v8f __attribute__((ext_vector_type(8)));

#define NQ    12
#define SDIM  4096
#define EPSV  1e-10f

// PennyLane qml.Rot entry U[r][c] for qubit q, weights w[q*3..q*3+2]=(phi,theta,omega)
static __device__ __forceinline__ void rot_entry(const float* __restrict__ w,
                                                 int q, int r, int c,
                                                 float& er, float& ei) {
  float phi = w[q * 3 + 0], th = w[q * 3 + 1], om = w[q * 3 + 2];
  float ch = cosf(0.5f * th), sh = sinf(0.5f * th);
  float hpo = 0.5f * (phi + om), hpm = 0.5f * (phi - om);
  float cpo = cosf(hpo), spo = sinf(hpo);
  float cpm = cosf(hpm), spm = sinf(hpm);
  if (r == 0) {
    if (c == 0) { er =  ch * cpo; ei = -ch * spo; }   // e^{-i(phi+om)/2} cos
    else        { er = -sh * cpm; ei = -sh * spm; }   // -e^{ i(phi-om)/2} sin
  } else {
    if (c == 0) { er =  sh * cpm; ei = -sh * spm; }   //  e^{ i(om-phi)/2} sin
    else        { er =  ch * cpo; ei =  ch * spo; }   //  e^{ i(phi+om)/2} cos
  }
}

// ws float layout: [0)A_re [4096)A_im [8192)B_re [12288)B_im ; then 4096 ints perm
// A and B both stored row-major [out][in].
__global__ void qdl_setup(const float* __restrict__ w,
                          float* __restrict__ mat,
                          int* __restrict__ perm) {
  int tid = blockIdx.x * blockDim.x + threadIdx.x;
  if (tid < 2 * SDIM) {
    int isB = tid >= SDIM;
    int idx = tid & (SDIM - 1);
    int outi = idx >> 6, ini = idx & 63;   // element [out][in]
    int qbase = isB ? 6 : 0;
    float re = 1.f, im = 0.f;
    for (int q = 0; q < 6; ++q) {
      int bo = (outi >> (5 - q)) & 1;      // qubit qbase+q is MSB-first in 6 bits
      int bi = (ini  >> (5 - q)) & 1;
      float er, ei;
      rot_entry(w, qbase + q, bo, bi, er, ei);
      float nre = re * er - im * ei;
      float nim = re * ei + im * er;
      re = nre; im = nim;
    }
    mat[(isB ? 8192 : 0) + idx]     = re;
    mat[(isB ? 12288 : 4096) + idx] = im;
  }
  if (tid < SDIM) {
    // out[m] = rot_state[perm[m]] after the CNOT cascade:
    // compose the (self-inverse) CNOT index maps in reverse application order.
    int src = tid;
    for (int i = NQ - 1; i >= 0; --i)
      for (int j = NQ - 1; j > i; --j)
        if ((src >> (NQ - 1 - i)) & 1) src ^= (1 << (NQ - 1 - j));
    perm[tid] = src;
  }
}

__global__ void __launch_bounds__(256)
qdl_main(const float* __restrict__ x, const float* __restrict__ mat,
         const int* __restrict__ perm, float* __restrict__ out) {
  __shared__ float buf0[SDIM];  // S_re  -> probabilities
  __shared__ float buf1[SDIM];  // T_re
  __shared__ float buf2[SDIM];  // T_im
  __shared__ float red[64];

  const int tid  = threadIdx.x;
  const int wave = tid >> 5;
  const int lane = tid & 31;
  const int ln15 = lane & 15;
  const int khalf = (lane >> 4) * 2;          // ISA 16x4 f32 A-frag K split
  const float* __restrict__ xr = x + (size_t)blockIdx.x * SDIM;

  // ---- Phase 1: load row, compute L2 norm, write normalized real state to LDS
  float vals[16];
  float ss = 0.f;
  for (int i = 0; i < 16; ++i) {
    float v = xr[tid + i * 256];
    vals[i] = v;
    ss += v * v;
  }
  for (int o = 16; o > 0; o >>= 1) ss += __shfl_xor(ss, o, 32);
  if (lane == 0) red[wave] = ss;
  __syncthreads();
  if (tid == 0) {
    float t = 0.f;
    for (int i = 0; i < 8; ++i) t += red[i];
    red[32] = t;
  }
  __syncthreads();
  float nrm = sqrtf(red[32]);
  bool zerovec = !(nrm > EPSV);
  float scale = zerovec ? 0.f : (1.0f / fmaxf(nrm, EPSV));
  for (int i = 0; i < 16; ++i)
    buf0[tid + i * 256] = zerovec ? 0.015625f : vals[i] * scale;  // 1/sqrt(4096)
  __syncthreads();

  const float* __restrict__ A_re = mat;
  const float* __restrict__ A_im = mat + 4096;
  const float* __restrict__ B_re = mat + 8192;   // row-major [n][k]
  const float* __restrict__ B_im = mat + 12288;

  // ---- Phase A: T = A * S  (S real)  -> buf1 (re), buf2 (im)
  // Each wave owns two 16x16 output tiles (same tm -> shared A fragments).
  for (int t = wave * 2; t < wave * 2 + 2; ++t) {
    int tm = t >> 2, tn = t & 3;
    int m = tm * 16 + ln15;
    int n = tn * 16 + ln15;
    v8f acc_re = {};
    v8f acc_im = {};
    for (int ks = 0; ks < 16; ++ks) {
      int k0 = ks * 4 + khalf;
      v2f ar, ai, bs;
      ar.x = A_re[m * 64 + k0];  ar.y = A_re[m * 64 + k0 + 1];
      ai.x = A_im[m * 64 + k0];  ai.y = A_im[m * 64 + k0 + 1];
      bs.x = buf0[k0 * 64 + n];  bs.y = buf0[(k0 + 1) * 64 + n];
      acc_re = __builtin_amdgcn_wmma_f32_16x16x4_f32(false, ar, false, bs,
                                                     (short)0, acc_re, false, false);
      acc_im = __builtin_amdgcn_wmma_f32_16x16x4_f32(false, ai, false, bs,
                                                     (short)0, acc_im, false, false);
    }
    int mrow = tm * 16 + (lane >> 4) * 8;     // ISA 16x16 f32 C/D layout
    for (int v = 0; v < 8; ++v) {
      buf1[(mrow + v) * 64 + n] = acc_re[v];
      buf2[(mrow + v) * 64 + n] = acc_im[v];
    }
  }
  __syncthreads();

  // ---- Phase B: S'' = T * B^T (complex); store |S''|^2 directly -> buf0
  // B-operand fragment for output col n over k-slice: B[n][k0..k0+1] (contiguous).
  for (int t = wave * 2; t < wave * 2 + 2; ++t) {
    int tm = t >> 2, tn = t & 3;
    int m = tm * 16 + ln15;
    int n = tn * 16 + ln15;
    v8f accP = {};  // sum T_re*B_re
    v8f accQ = {};  // sum T_im*B_im   (re = P - Q)
    v8f accI = {};  // sum T_re*B_im + T_im*B_re
    for (int ks = 0; ks < 16; ++ks) {
      int k0 = ks * 4 + khalf;
      v2f tre, tim, bre, bim;
      tre.x = buf1[m * 64 + k0];   tre.y = buf1[m * 64 + k0 + 1];
      tim.x = buf2[m * 64 + k0];   tim.y = buf2[m * 64 + k0 + 1];
      bre.x = B_re[n * 64 + k0];   bre.y = B_re[n * 64 + k0 + 1];
      bim.x = B_im[n * 64 + k0];   bim.y = B_im[n * 64 + k0 + 1];
      accP = __builtin_amdgcn_wmma_f32_16x16x4_f32(false, tre, false, bre,
                                                   (short)0, accP, false, false);
      accQ = __builtin_amdgcn_wmma_f32_16x16x4_f32(false, tim, false, bim,
                                                   (short)0, accQ, false, false);
      accI = __builtin_amdgcn_wmma_f32_16x16x4_f32(false, tre, false, bim,
                                                   (short)0, accI, false, false);
      accI = __builtin_amdgcn_wmma_f32_16x16x4_f32(false, tim, false, bre,
                                                   (short)0, accI, false, false);
    }
    int mrow = tm * 16 + (lane >> 4) * 8;
    for (int v = 0; v < 8; ++v) {
      float re = accP[v] - accQ[v];
      float im = accI[v];
      buf0[(mrow + v) * 64 + n] = re * re + im * im;
    }
  }
  __syncthreads();

  // ---- Phase C: normalize probabilities, apply CNOT permutation on store
  float psum = 0.f;
  for (int i = 0; i < 16; ++i) psum += buf0[tid + i * 256];
  for (int o = 16; o > 0; o >>= 1) psum += __shfl_xor(psum, o, 32);
  if (lane == 0) red[wave] = psum;
  __syncthreads();
  if (tid == 0) {
    float t = 0.f;
    for (int i = 0; i < 8; ++i) t += red[i];
    red[32] = t;
  }
  __syncthreads();
  float ps = red[32];
  bool zp = !(ps > EPSV);
  float inv = zp ? 0.f : (1.0f / fmaxf(ps, EPSV));
  float* __restrict__ orow = out + (size_t)blockIdx.x * SDIM;
  for (int i = 0; i < 16; ++i) {
    int m = tid + i * 256;
    orow[m] = zp ? (1.0f / 4096.0f) : buf0[perm[m]] * inv;
  }
}

extern "C" void kernel_launch(void* const* d_in, const int* in_sizes, int n_in,
                              void* d_out, int out_size, void* d_ws, size_t ws_size,
                              hipStream_t stream) {
  const float* x = (const float*)d_in[0];          // (2048, 4096) f32
  const float* w = (const float*)d_in[1];          // (1, 12, 3)   f32
  float* out = (float*)d_out;                      // (2048, 4096) f32
  float* mat = (float*)d_ws;                       // 16384 floats
  int* perm = (int*)(mat + 16384);                 // 4096 ints (80 KB total ws use)

  qdl_setup<<<32, 256, 0, stream>>>(w, mat, perm);

  int batch = in_sizes[0] / SDIM;                  // 2048
  qdl_main<<<batch, 256, 0, stream>>>(x, mat, perm, out);
}